// loss_SAW_1_83863531421939
// MI455X (gfx1250) — compile-verified
//
#include <hip/hip_runtime.h>
#include <hip/hip_bf16.h>

// loss_SAW: segmented variance-style loss over 4096x4096 image with 65 mask ids.
// Memory-bound: 192MB traffic -> ~8.2us floor at 23.3 TB/s on MI455X.
// Plan:
//   stage1: stream float4/int4, v_log_f32 for ln(), per-wave LDS histograms via
//           ds_add_f32 atomics, each WG writes a private 208-float partial row.
//   stage2: reduce G partial rows with V_WMMA_F32_16X16X4_F32 (A = ones), full
//           f32 precision, 13 waves (one per 16-column tile), then finalize.

#define EPSF 1e-6f
#define NSEG 65          // mask ids 0..64 (segment 0 unused in the final sum)
#define NSTAT 3          // sum, sumsq, count
#define PJ   208         // 195 live columns (3*65) padded to 13 WMMA tiles of 16
#define WGSZ 256
#define NWAVES (WGSZ / 32)

typedef __attribute__((ext_vector_type(2))) float v2f;
typedef __attribute__((ext_vector_type(8))) float v8f;

// ---------------------------------------------------------------- stage 1 ----
__global__ __launch_bounds__(WGSZ) void
saw_stage1(const float* __restrict__ shading,
           const float* __restrict__ image,
           const int*   __restrict__ mask,
           float*       __restrict__ P,      // [G][PJ] partials
           int n, int G)
{
    __shared__ float hist[NWAVES * PJ];
    for (int j = threadIdx.x; j < NWAVES * PJ; j += WGSZ) hist[j] = 0.0f;
    __syncthreads();

    float* h = &hist[(threadIdx.x >> 5) * PJ];   // per-wave histogram

    const float4* s4 = (const float4*)shading;
    const float4* i4 = (const float4*)image;
    const int4*   m4 = (const int4*)mask;
    const int n4     = n >> 2;
    const int gid    = blockIdx.x * WGSZ + threadIdx.x;
    const int stride = G * WGSZ;

    for (int i = gid; i < n4; i += stride) {
        float4 sv = s4[i];
        float4 iv = i4[i];
        int4   mv = m4[i];

        float s[4] = {sv.x, sv.y, sv.z, sv.w};
        float v[4] = {iv.x, iv.y, iv.z, iv.w};
        int   m[4] = {mv.x, mv.y, mv.z, mv.w};

#pragma unroll
        for (int c = 0; c < 4; ++c) {
            // ln via v_log_f32 (log2) * ln2 — fast transcendental path
            float ld  = __logf(s[c] + EPSF) - __logf(v[c] + EPSF);
            int   seg = m[c];
            atomicAdd(&h[seg],            ld);        // ds_add_f32
            atomicAdd(&h[NSEG + seg],     ld * ld);   // ds_add_f32
            atomicAdd(&h[2 * NSEG + seg], 1.0f);      // ds_add_f32 (exact < 2^24)
        }
    }
    __syncthreads();

    // Fold the per-wave histograms and emit this WG's partial row (pads are 0).
    for (int j = threadIdx.x; j < PJ; j += WGSZ) {
        float acc = 0.0f;
#pragma unroll
        for (int w = 0; w < NWAVES; ++w) acc += hist[w * PJ + j];
        P[(size_t)blockIdx.x * PJ + j] = acc;
    }
}

// ---------------------------------------------------------------- stage 2 ----
// 13 waves; wave w reduces columns [16w, 16w+16) of P over all G rows using
// D += ones(16x4) * B(4x16) f32 WMMA accumulation (exact: f32 MACs, and the
// all-ones A makes the result independent of the K-row ordering inside B).
__global__ __launch_bounds__(416) void
saw_stage2(const float* __restrict__ P,
           float*       __restrict__ out,
           const int*   __restrict__ numMaskPtr,
           float num, int G)
{
    __shared__ float segsum[PJ];

    const int lane = threadIdx.x & 31;
    const int wave = threadIdx.x >> 5;            // 0..12
    const int col  = wave * 16 + (lane & 15);     // column this lane carries
    const int krow = (lane >> 4) * 2;             // B rows: lanes 0-15 -> K0,K1; 16-31 -> K2,K3

    v2f a = {1.0f, 1.0f};                         // A = all-ones (16x4)
    v8f c = {};                                   // D accumulator (16x16 f32)

    for (int g0 = 0; g0 < G; g0 += 4) {
        v2f b;
        b.x = P[(size_t)(g0 + krow)     * PJ + col];
        b.y = P[(size_t)(g0 + krow + 1) * PJ + col];
        c = __builtin_amdgcn_wmma_f32_16x16x4_f32(
                /*neg_a=*/false, a, /*neg_b=*/false, b,
                /*c_mod=*/(short)0, c, /*reuse_a=*/false, /*reuse_b=*/false);
    }

    // D row M=0 lives in VGPR c[0] of lanes 0..15 (all rows are identical).
    if (lane < 16) segsum[col] = c[0];
    __syncthreads();

    if (threadIdx.x == 0) {
        int nm = numMaskPtr[0];
        if (nm > NSEG - 1) nm = NSEG - 1;
        float total = 0.0f;
        for (int s = 1; s <= nm; ++s) {
            float ssum = segsum[s];
            float ssq  = segsum[NSEG + s];
            float cnt  = segsum[2 * NSEG + s];
            float Nn   = cnt * num;
            float s1   = ssq / (Nn + EPSF);
            float s2   = (ssum * ssum) / (Nn * Nn + EPSF);
            total += (s1 - s2);
        }
        out[0] = total / ((float)nm + EPSF);
    }
}

// ----------------------------------------------------------------- launch ----
extern "C" void kernel_launch(void* const* d_in, const int* in_sizes, int n_in,
                              void* d_out, int out_size, void* d_ws, size_t ws_size,
                              hipStream_t stream)
{
    const float* shading = (const float*)d_in[0];
    const float* image   = (const float*)d_in[1];
    const int*   mask    = (const int*)d_in[2];
    const int*   numMask = (const int*)d_in[3];
    float*       out     = (float*)d_out;
    float*       P       = (float*)d_ws;

    const int n = in_sizes[0];                    // 4096*4096

    // Size stage-1 grid to the workspace (832 B per WG row), multiple of 4
    // rows so stage 2's K=4 chunking is exact.
    const size_t rowBytes = (size_t)PJ * sizeof(float);
    long long g = (long long)(ws_size / rowBytes);
    if (g > 1024) g = 1024;
    g &= ~3LL;
    if (g < 4) g = 4;
    const int G = (int)g;

    saw_stage1<<<G, WGSZ, 0, stream>>>(shading, image, mask, P, n, G);
    saw_stage2<<<1, 416, 0, stream>>>(P, out, numMask, (float)n, G);
}